// EchoSpike_layer_35957466202608
// MI455X (gfx1250) — compile-verified
//
#include <hip/hip_runtime.h>

// ---------------------------------------------------------------------------
// EchoSpike layer, MI455X (gfx1250).
// Dominant cost: current_dW update = 512MB read + 512MB write ~= 1GB
//   -> memory-bound, ~44us at 23.3 TB/s. Everything else is noise.
// GEMM (128x1024x1024, 0.27 GFLOP) done with native f32 WMMA
//   (v_wmma_f32_16x16x4_f32) for exact f32 fidelity, fused with the LIF step.
// ---------------------------------------------------------------------------

#define BDIM   128
#define NIN    1024
#define NH     1024
#define BETA   0.95f
#define INV_T  0.01f               // 1 / N_TIME_STEPS
#define PI_F   3.14159265358979323846f

// Output layout (flat f32 elements, reference tuple order):
#define OFF_SPK  0                  // (B,NH)
#define OFF_MEM  131072             // (B,NH)
#define OFF_LOSS 262144             // scalar  <- shifts everything after by 1!
#define OFF_STR  262145             // (B,NH)  spk_trace_new
#define OFF_ITR  393217             // (B,NIN) inp_trace_new
#define OFF_DW   524289             // (B,NH,NIN), start == 1 (mod 4)
#define OFF_SL   134742017u         // (B,)

typedef __attribute__((ext_vector_type(2))) float v2f;
typedef __attribute__((ext_vector_type(4))) float v4f;
typedef __attribute__((ext_vector_type(8))) float v8f;

// ---------------------------------------------------------------------------
// Kernel A: cur = inp @ W^T via V_WMMA_F32_16X16X4_F32, fused LIF epilogue.
// 8 waves/block, one 16x16 output tile per wave. 512 tiles -> 64 blocks.
// A-layout (16x4 f32): lanes 0-15 hold K=0,1 in v0,v1; lanes 16-31 hold K=2,3.
// B-layout (4x16 f32): lane n holds B[k][n] = W[hbase+n][k]; same half split.
// ---------------------------------------------------------------------------
__global__ __launch_bounds__(256) void k_gemm_lif(
    const float* __restrict__ inp, const float* __restrict__ W,
    const float* __restrict__ mem, const float* __restrict__ spk_trace,
    float* __restrict__ out)
{
    const int wave = threadIdx.x >> 5;
    const int lane = threadIdx.x & 31;
    const int tile = blockIdx.x * 8 + wave;   // 0..511
    const int bt   = tile >> 6;               // batch tile   0..7
    const int ht   = tile & 63;               // hidden tile  0..63
    const int half = lane >> 4;               // 0: K 0-1 / M 0-7, 1: K 2-3 / M 8-15
    const int r16  = lane & 15;               // A row (M) and B col (N)
    const int koff = half << 1;

    const float* pa = inp + (size_t)(bt * 16 + r16) * NIN + koff;
    const float* pb = W   + (size_t)(ht * 16 + r16) * NIN + koff;

    v8f acc = {0.f, 0.f, 0.f, 0.f, 0.f, 0.f, 0.f, 0.f};
    for (int k = 0; k < NIN; k += 16) {
#pragma unroll
        for (int kk = 0; kk < 16; kk += 4) {
            v2f a = *(const v2f*)(pa + k + kk);
            v2f b = *(const v2f*)(pb + k + kk);
            acc = __builtin_amdgcn_wmma_f32_16x16x4_f32(
                false, a, false, b, (short)0, acc, false, false);
        }
    }

    // Epilogue: C/D layout -> element (M = r + 8*half, N = r16) in acc[r].
    const int h  = ht * 16 + r16;
    const int bb = bt * 16 + (half << 3);
#pragma unroll
    for (int r = 0; r < 8; ++r) {
        const size_t idx = (size_t)(bb + r) * NH + h;
        const float cur   = acc[r];
        const float m0    = mem[idx];
        const float reset = (m0 > 1.0f) ? 1.0f : 0.0f;
        const float mn    = BETA * m0 + cur - reset;   // reset-by-subtraction
        const float spk   = (mn > 1.0f) ? 1.0f : 0.0f;
        out[OFF_SPK + idx] = spk;
        out[OFF_MEM + idx] = mn;
        out[OFF_STR + idx] = spk_trace[idx] + spk * INV_T;
    }
}

// ---------------------------------------------------------------------------
// Kernel B: inp_trace_new = 0.95*inp_trace + inp   (B*NIN = 131072 elems)
// ---------------------------------------------------------------------------
__global__ void k_inp_trace(const float* __restrict__ inp,
                            const float* __restrict__ itr,
                            float* __restrict__ out)
{
    const int i = blockIdx.x * blockDim.x + threadIdx.x;
    if (i < BDIM * NIN) out[OFF_ITR + i] = 0.95f * itr[i] + inp[i];
}

// ---------------------------------------------------------------------------
// Kernel C: per-sample loss = -bf * sum_h spk*prev ; sample_loss_new = +loss
// ---------------------------------------------------------------------------
__global__ __launch_bounds__(256) void k_loss(
    const float* __restrict__ prev, const float* __restrict__ sample_loss,
    const int* __restrict__ bf, float* __restrict__ out,
    float* __restrict__ ws_loss)
{
    __shared__ float red[256];
    const int b = blockIdx.x, t = threadIdx.x;
    const float* s = out + OFF_SPK + (size_t)b * NH;
    const float* p = prev + (size_t)b * NH;
    float acc = 0.0f;
    for (int h = t; h < NH; h += 256) acc += s[h] * p[h];
    red[t] = acc;
    __syncthreads();
    for (int o = 128; o > 0; o >>= 1) {
        if (t < o) red[t] += red[t + o];
        __syncthreads();
    }
    if (t == 0) {
        const float loss = -(float)bf[0] * red[0];
        ws_loss[b] = loss;
        out[OFF_SL + b] = sample_loss[b] + loss;
    }
}

// Kernel D: mean of per-sample losses -> scalar slot
__global__ void k_mean(const float* __restrict__ ws_loss, float* __restrict__ out)
{
    __shared__ float red[128];
    const int t = threadIdx.x;
    red[t] = ws_loss[t];
    __syncthreads();
    for (int o = 64; o > 0; o >>= 1) {
        if (t < o) red[t] += red[t + o];
        __syncthreads();
    }
    if (t == 0) out[OFF_LOSS] = red[0] * (1.0f / BDIM);
}

// ---------------------------------------------------------------------------
// Kernel E: the 1 GB stream.  dW_new = dW + bf * pre(b,h) * inp_trace_new(b,i)
// One block per (b,h) row (131072 blocks x 256 threads, 4 floats/thread).
// Input rows are 16B aligned; output rows start at global elem ==1 (mod 4)
// because of the scalar loss slot -> realign through LDS so BOTH the 512MB
// read and the 512MB write use aligned b128 non-temporal ops (bypass-friendly:
// 1GB stream >> 192MB L2).  pre and inp_trace_new recomputed from small
// L2-resident inputs (512KB each) instead of round-tripping scratch.
// ---------------------------------------------------------------------------
__global__ __launch_bounds__(256) void k_dw(
    const float* __restrict__ dW_in, const float* __restrict__ inp,
    const float* __restrict__ itr_in, const float* __restrict__ prev,
    const int* __restrict__ bf, float* __restrict__ out)
{
    __shared__ float lds[NIN];
    const int row = blockIdx.x;          // b*NH + h
    const int b   = row >> 10;
    const int t   = threadIdx.x;

    // pre = prev_spk_trace * surrogate(mem_new - 1), uniform per row
    const float mn  = out[OFF_MEM + row];
    const float pv  = prev[row];
    const float x   = mn - 1.0f;
    const float px  = PI_F * x;
    const float pre = pv / (PI_F * (1.0f + px * px));
    const float s   = (float)bf[0] * pre;

    const v4f* vin = (const v4f*)(dW_in + (size_t)row * NIN);
    const v4f* vit = (const v4f*)(itr_in + (size_t)b * NIN);
    const v4f* vsp = (const v4f*)(inp    + (size_t)b * NIN);

    const v4f w  = __builtin_nontemporal_load(vin + t);   // aligned b128 NT
    const v4f it = 0.95f * vit[t] + vsp[t];               // inp_trace_new
    const v4f u  = w + s * it;

    const int j4 = t << 2;
    lds[j4 + 0] = u.x; lds[j4 + 1] = u.y; lds[j4 + 2] = u.z; lds[j4 + 3] = u.w;
    __syncthreads();

    float* orow = out + OFF_DW + (size_t)row * NIN;   // row base == 1 (mod 4)
    if (t < 255) {
        const int j = j4 + 3;                         // ==0 (mod 4) in out frame
        v4f v;
        v.x = lds[j]; v.y = lds[j + 1]; v.z = lds[j + 2]; v.w = lds[j + 3];
        __builtin_nontemporal_store(v, (v4f*)(orow + j));   // aligned b128 NT
    } else {
        __builtin_nontemporal_store(lds[0], orow + 0);
        __builtin_nontemporal_store(lds[1], orow + 1);
        __builtin_nontemporal_store(lds[2], orow + 2);
        __builtin_nontemporal_store(lds[NIN - 1], orow + NIN - 1);
    }
}

// ---------------------------------------------------------------------------
extern "C" void kernel_launch(void* const* d_in, const int* in_sizes, int n_in,
                              void* d_out, int out_size, void* d_ws, size_t ws_size,
                              hipStream_t stream)
{
    const float* inp         = (const float*)d_in[0];
    const float* W           = (const float*)d_in[1];
    const float* mem         = (const float*)d_in[2];
    const float* spk_trace   = (const float*)d_in[3];
    const float* inp_trace   = (const float*)d_in[4];
    const float* prev        = (const float*)d_in[5];
    const float* dW          = (const float*)d_in[6];
    const float* sample_loss = (const float*)d_in[7];
    const int*   bf          = (const int*)d_in[8];
    float* out     = (float*)d_out;
    float* ws_loss = (float*)d_ws;   // 128 floats

    k_gemm_lif <<<64,          256, 0, stream>>>(inp, W, mem, spk_trace, out);
    k_inp_trace<<<512,         256, 0, stream>>>(inp, inp_trace, out);
    k_loss     <<<BDIM,        256, 0, stream>>>(prev, sample_loss, bf, out, ws_loss);
    k_mean     <<<1,           128, 0, stream>>>(ws_loss, out);
    k_dw       <<<BDIM * NH,   256, 0, stream>>>(dW, inp, inp_trace, prev, bf, out);
}